// Split_56281251447201
// MI455X (gfx1250) — compile-verified
//
#include <hip/hip_runtime.h>

#define B_ 32
#define S_ 1024
#define E_ 128
#define N_ 17          // 1 + ADDITIONAL
#define BS_ 256        // S >> DEPTH
#define NB_ 4          // S / BS
#define NPAD_ 20
#define GSTRIDE_ 132   // padded LDS row stride (floats)

typedef __attribute__((ext_vector_type(2))) float v2f;
typedef __attribute__((ext_vector_type(8))) float v8f;

// ---- CDNA5 async memory->LDS copy (16B per lane), tracked by ASYNCcnt --------
__device__ __forceinline__ void async_ld16(unsigned lds_addr, unsigned long long gaddr) {
  asm volatile("global_load_async_to_lds_b128 %0, %1, off"
               :: "v"(lds_addr), "v"(gaddr) : "memory");
}
__device__ __forceinline__ void wait_async0() {
#if __has_builtin(__builtin_amdgcn_s_wait_asynccnt)
  __builtin_amdgcn_s_wait_asynccnt(0);
#else
  asm volatile("s_wait_asynccnt 0x0" ::: "memory");
#endif
}

// ---------------- Kernel 1: stable-partition targets -> inverse permutation ----
__global__ __launch_bounds__(BS_) void k_split_inv(const float* __restrict__ offset,
                                                   const int* __restrict__ sampled,
                                                   int* __restrict__ inv) {
  int gid = blockIdx.x;
  int blk = gid % NB_; gid /= NB_;
  int j   = gid % N_;  gid /= N_;
  int b   = gid;
  int t = threadIdx.x;
  int s = blk * BS_ + t;
  int c;
  if (j == 0) c = offset[b * S_ + s] > 0.5f;                       // round-half-even on [0,1)
  else        c = sampled[((size_t)b * (N_ - 1) + (j - 1)) * S_ + s] != 0;
  __shared__ int sc[BS_];
  sc[t] = c;
  __syncthreads();
  for (int off = 1; off < BS_; off <<= 1) {                        // inclusive scan
    int v = (t >= off) ? sc[t - off] : 0;
    __syncthreads();
    sc[t] += v;
    __syncthreads();
  }
  int ones_before  = sc[t] - c;
  int total_ones   = sc[BS_ - 1];
  int zeros_before = t - ones_before;
  int nzeros       = BS_ - total_ones;
  int tgt = c ? (nzeros + ones_before) : zeros_before;
  inv[((size_t)b * N_ + j) * S_ + blk * BS_ + tgt] = s;            // permutation: no conflicts
}

// ---------------- Kernel 2: raw choice-vector probabilities (f32 product) ------
__global__ __launch_bounds__(256) void k_probs(const float* __restrict__ offset,
                                               const int* __restrict__ sampled,
                                               float* __restrict__ praw) {
  int j = blockIdx.x % N_, b = blockIdx.x / N_;
  int t = threadIdx.x;
  float p = 1.0f;
  for (int k = 0; k < S_ / 256; ++k) {
    int s = t + k * 256;
    float o = offset[b * S_ + s];
    int c = (j == 0) ? (o > 0.5f)
                     : (sampled[((size_t)b * (N_ - 1) + (j - 1)) * S_ + s] != 0);
    p *= c ? o : (1.0f - o);
  }
  __shared__ float red[256];
  red[t] = p; __syncthreads();
  for (int off = 128; off > 0; off >>= 1) { if (t < off) red[t] *= red[t + off]; __syncthreads(); }
  if (t == 0) praw[b * N_ + j] = red[0];
}

// ---------------- Kernel 3: duplicate-permutation mask -------------------------
__global__ __launch_bounds__(256) void k_dup(const int* __restrict__ inv,
                                             int* __restrict__ dup) {
  int j = blockIdx.x % N_, b = blockIdx.x / N_;
  int t = threadIdx.x;
  __shared__ int red[256];
  int isdup = 0;
  for (int j2 = 0; j2 < j; ++j2) {
    int eq = 1;
    for (int k = 0; k < S_ / 256; ++k) {
      int s = t + k * 256;
      eq &= (inv[((size_t)b * N_ + j ) * S_ + s] ==
             inv[((size_t)b * N_ + j2) * S_ + s]);
    }
    red[t] = eq; __syncthreads();
    for (int off = 128; off > 0; off >>= 1) { if (t < off) red[t] &= red[t + off]; __syncthreads(); }
    isdup |= red[0];
    __syncthreads();
  }
  if (t == 0) dup[b * N_ + j] = isdup;
}

// ---------------- Kernel 4: mask + normalize (wave32 shuffle reduce) -----------
__global__ __launch_bounds__(32) void k_norm(const float* __restrict__ praw,
                                             const int* __restrict__ dup,
                                             float* __restrict__ ppad) {
  int b = blockIdx.x, j = threadIdx.x;
  float p = 0.0f;
  if (j < N_) { p = praw[b * N_ + j]; if (dup[b * N_ + j]) p = 0.0f; }
  float sum = p;
  #pragma unroll
  for (int off = 16; off > 0; off >>= 1) sum += __shfl_xor(sum, off, 32);
  float pn = p / sum;                               // 0/0 -> NaN, matching reference f32 path
  if (j < NPAD_) ppad[b * NPAD_ + j] = (j < N_) ? pn : 0.0f;
}

// ---------------- Kernel 5: WMMA combine with async-LDS double buffering -------
// out[r,:] = sum_j p_j * input[inv_j(r),:]
__global__ __launch_bounds__(128) void k_apply_wmma(const float* __restrict__ input,
                                                    const int* __restrict__ inv,
                                                    const float* __restrict__ ppad,
                                                    float* __restrict__ out) {
  int rt = blockIdx.x;           // row tile 0..63
  int b  = blockIdx.y;           // batch
  int t    = threadIdx.x;
  int lane = t & 31;
  int w    = t >> 5;             // wave 0..3 -> E-tiles 2w, 2w+1
  __shared__ __align__(16) float Gs[2][16 * GSTRIDE_];
  __shared__ int   inv_s[N_ * 16];
  __shared__ float pn_s[NPAD_];
  for (int i = t; i < N_ * 16; i += 128) {
    int jj = i >> 4, rr = i & 15;
    inv_s[i] = inv[((size_t)b * N_ + jj) * S_ + rt * 16 + rr];
  }
  if (t < NPAD_) pn_s[t] = ppad[b * NPAD_ + t];
  v8f acc0 = {}; v8f acc1 = {};
  int m    = lane & 15;          // D column / A row
  int hi   = lane >> 4;          // lane half
  int kkb  = hi * 2;             // K slots {0,1} or {2,3} per documented f32 layout
  int row8 = t >> 3;             // staging: 8 threads per row
  int quad = t & 7;              // 4x16B per thread, stride 128B
  __syncthreads();

  // per-thread fixed LDS byte offsets for the two buffers (low 32 bits of the
  // generic pointer = wave-relative LDS address used by async VDST)
  unsigned lds0 = (unsigned)(unsigned long long)(const void*)
                  &Gs[0][row8 * GSTRIDE_ + quad * 4];
  unsigned lds1 = (unsigned)(unsigned long long)(const void*)
                  &Gs[1][row8 * GSTRIDE_ + quad * 4];

  // stage j=0 into buffer 0
  {
    int src = inv_s[0 * 16 + row8];
    unsigned long long ga = (unsigned long long)(const void*)
                            (input + ((size_t)b * S_ + src) * E_ + quad * 4);
    #pragma unroll
    for (int k = 0; k < 4; ++k) async_ld16(lds0 + k * 128u, ga + k * 128u);
  }
  wait_async0();
  __syncthreads();

  for (int j = 0; j < N_; ++j) {
    int buf = j & 1;
    if (j + 1 < N_) {            // prefetch next gathered tile into idle buffer
      int src = inv_s[(j + 1) * 16 + row8];
      unsigned long long ga = (unsigned long long)(const void*)
                              (input + ((size_t)b * S_ + src) * E_ + quad * 4);
      unsigned lds = buf ? lds0 : lds1;
      #pragma unroll
      for (int k = 0; k < 4; ++k) async_ld16(lds + k * 128u, ga + k * 128u);
    }
    float pj = pn_s[j];
    const float* G = &Gs[buf][0];
    #pragma unroll
    for (int ks = 0; ks < 4; ++ks) {             // K chunk: block-diag A, rows of G as B
      int rl = ks * 4 + kkb;
      v2f a;                                      // A[m,k] = p_j * delta(m == k-row)
      a.x = (m == rl    ) ? pj : 0.0f;
      a.y = (m == rl + 1) ? pj : 0.0f;
      {
        int e0 = (w * 2 + 0) * 16;
        v2f bb;
        bb.x = G[ rl      * GSTRIDE_ + e0 + m];
        bb.y = G[(rl + 1) * GSTRIDE_ + e0 + m];
        acc0 = __builtin_amdgcn_wmma_f32_16x16x4_f32(false, a, false, bb,
                                                     (short)0, acc0, false, false);
      }
      {
        int e0 = (w * 2 + 1) * 16;
        v2f bb;
        bb.x = G[ rl      * GSTRIDE_ + e0 + m];
        bb.y = G[(rl + 1) * GSTRIDE_ + e0 + m];
        acc1 = __builtin_amdgcn_wmma_f32_16x16x4_f32(false, a, false, bb,
                                                     (short)0, acc1, false, false);
      }
    }
    wait_async0();               // own async writes (into idle buffer) complete
    __syncthreads();             // all waves done reading buf -> reusable next iter
  }

  // store D per documented 16x16 f32 C/D layout: VGPR v -> M = v + 8*hi, N = lane&15
  size_t base = ((size_t)b * S_ + rt * 16) * E_;
  #pragma unroll
  for (int v = 0; v < 8; ++v) {
    int mr = v + hi * 8;
    out[base + (size_t)mr * E_ + (w * 2 + 0) * 16 + m] = acc0[v];
    out[base + (size_t)mr * E_ + (w * 2 + 1) * 16 + m] = acc1[v];
  }
}

// ---------------- Kernel 6: keys combine ---------------------------------------
__global__ __launch_bounds__(256) void k_keys(const float* __restrict__ keys,
                                              const int* __restrict__ inv,
                                              const float* __restrict__ ppad,
                                              float* __restrict__ keys_out) {
  int idx = blockIdx.x * 256 + threadIdx.x;
  int b = idx / S_, r = idx % S_;
  float acc = 0.0f;
  for (int j = 0; j < N_; ++j) {
    int src = inv[((size_t)b * N_ + j) * S_ + r];
    acc += ppad[b * NPAD_ + j] * keys[b * S_ + src];
  }
  keys_out[idx] = acc;
}

extern "C" void kernel_launch(void* const* d_in, const int* in_sizes, int n_in,
                              void* d_out, int out_size, void* d_ws, size_t ws_size,
                              hipStream_t stream) {
  const float* input   = (const float*)d_in[0];
  const float* keys    = (const float*)d_in[1];
  const float* offset  = (const float*)d_in[2];
  const int*   sampled = (const int*)d_in[3];
  float* out      = (float*)d_out;
  float* keys_out = out + (size_t)B_ * S_ * E_;

  // workspace layout (~2.24 MB)
  int*   inv  = (int*)d_ws;                               // B*N*S ints
  float* praw = (float*)(inv + (size_t)B_ * N_ * S_);     // B*N
  int*   dup  = (int*)(praw + B_ * N_);                   // B*N
  float* ppad = (float*)(dup + B_ * N_);                  // B*NPAD

  k_split_inv <<<dim3(B_ * N_ * NB_), dim3(BS_), 0, stream>>>(offset, sampled, inv);
  k_probs     <<<dim3(B_ * N_),       dim3(256), 0, stream>>>(offset, sampled, praw);
  k_dup       <<<dim3(B_ * N_),       dim3(256), 0, stream>>>(inv, dup);
  k_norm      <<<dim3(B_),            dim3(32),  0, stream>>>(praw, dup, ppad);
  k_apply_wmma<<<dim3(S_ / 16, B_),   dim3(128), 0, stream>>>(input, inv, ppad, out);
  k_keys      <<<dim3(B_ * S_ / 256), dim3(256), 0, stream>>>(keys, inv, ppad, keys_out);
}